// Net_54614804136391
// MI455X (gfx1250) — compile-verified
//
#include <hip/hip_runtime.h>
#include <hip/hip_bf16.h>
#include <math.h>

typedef __attribute__((ext_vector_type(2))) float v2f;
typedef __attribute__((ext_vector_type(8))) float v8f;

#define HH 256
#define WW 256
#define KPTS 512
#define CDIM 192
#define C1DIM 64
#define C2DIM 128
#define H2 128
#define W2 128
#define NPIX 65536
#define BORDER_PX 10
#define NSLOTS 4 /* (img,b) pairs: img*2+b */

/* d_out float offsets (return order: probs1, probs2, lm1, lm2, score, norm) */
#define PROBS1_OFF 0
#define PROBS2_OFF 1024
#define LM1_OFF 2048
#define LM2_OFF 4096
#define SCORE_OFF 6144
#define NORM_OFF (6144 + 2 * KPTS * KPTS * 2) /* 1054720 */

// ---------------------------------------------------------------------------
// Kernel 1: sigmoid + 8x8 block-max NMS + border mask.
// One 64-thread block per 8x8 window.
// ---------------------------------------------------------------------------
__global__ void nms_kernel(const float* __restrict__ heat1,
                           const float* __restrict__ heat2,
                           float* __restrict__ nms) {
  int slot = blockIdx.y;            // img*2 + b
  int img = slot >> 1, b = slot & 1;
  const float* heat = (img ? heat2 : heat1) + (size_t)b * NPIX;
  float* out = nms + (size_t)slot * NPIX;
  int win = blockIdx.x;             // 0..1023
  int wy = (win >> 5) << 3;
  int wx = (win & 31) << 3;
  int t = threadIdx.x;              // 0..63
  int r = wy + (t >> 3);
  int c = wx + (t & 7);
  float h = 1.0f / (1.0f + __expf(-heat[r * WW + c]));
  __shared__ float sm[64];
  sm[t] = h;
  __syncthreads();
  for (int off = 32; off > 0; off >>= 1) {
    if (t < off) sm[t] = fmaxf(sm[t], sm[t + off]);
    __syncthreads();
  }
  float bmax = sm[0];
  bool keep = (h >= bmax) &&
              (r >= BORDER_PX) && (r < HH - BORDER_PX) &&
              (c >= BORDER_PX) && (c < WW - BORDER_PX);
  out[r * WW + c] = keep ? h : 0.0f;
}

// ---------------------------------------------------------------------------
// Kernel 2: exact top-512 per image via 8-bit radix select on value bits
// (all values >= 0 so uint order == float order), index radix for ties,
// then 512-element bitonic sort in LDS for (value desc, index asc) order.
// One 1024-thread block per image slot.
// ---------------------------------------------------------------------------
__global__ __launch_bounds__(1024) void topk_kernel(
    const float* __restrict__ nms, int* __restrict__ topkIdx) {
  int slot = blockIdx.x;
  const float* base = nms + (size_t)slot * NPIX;
  int tid = threadIdx.x;
  __shared__ unsigned bins[256];
  __shared__ unsigned s_sel, s_need, s_neq, s_cnt;
  __shared__ unsigned s_prefix;
  __shared__ unsigned long long keys[KPTS];

  unsigned prefix = 0, pmask = 0, need = KPTS;
  for (int shift = 24; shift >= 0; shift -= 8) {
    if (tid < 256) bins[tid] = 0;
    __syncthreads();
    for (int e = tid; e < NPIX; e += 1024) {
      unsigned u = __float_as_uint(base[e]);
      if ((u & pmask) == prefix) atomicAdd(&bins[(u >> shift) & 255u], 1u);
    }
    __syncthreads();
    if (tid == 0) {
      unsigned cum = 0; int sel = 0;
      for (int i = 255; i >= 0; --i) {
        cum += bins[i];
        if (cum >= need) { sel = i; break; }
      }
      unsigned above = cum - bins[sel];
      s_sel = (unsigned)sel;
      s_need = need - above;        // how many equal-or-below in this bin
      s_neq = bins[sel];
      s_prefix = prefix | ((unsigned)sel << shift);
    }
    __syncthreads();
    prefix = s_prefix;
    need = s_need;
    pmask |= (255u << shift);
    __syncthreads();
  }
  unsigned V = prefix;              // bit pattern of the 512th largest value
  unsigned neq = s_neq;             // count of elements == V
  unsigned cutIdx = 0xFFFFu;        // take equals with idx <= cutIdx
  if (need < neq) {
    // ascending radix select on 16-bit flat index among elements == V
    if (tid < 256) bins[tid] = 0;
    __syncthreads();
    for (int e = tid; e < NPIX; e += 1024) {
      if (__float_as_uint(base[e]) == V) atomicAdd(&bins[(e >> 8) & 255], 1u);
    }
    __syncthreads();
    if (tid == 0) {
      unsigned cum = 0; int sel = 0;
      for (int i = 0; i < 256; ++i) {
        if (cum + bins[i] >= need) { sel = i; break; }
        cum += bins[i];
      }
      s_sel = (unsigned)sel;
      s_need = need - cum;
    }
    __syncthreads();
    unsigned hb = s_sel, need2 = s_need;
    if (tid < 256) bins[tid] = 0;
    __syncthreads();
    for (int e = tid; e < NPIX; e += 1024) {
      if (__float_as_uint(base[e]) == V && (((unsigned)e >> 8) & 255u) == hb)
        atomicAdd(&bins[e & 255], 1u);
    }
    __syncthreads();
    if (tid == 0) {
      unsigned cum = 0; int sel = 0;
      for (int i = 0; i < 256; ++i) {
        cum += bins[i];
        if (cum >= need2) { sel = i; break; }
      }
      s_sel = (unsigned)sel;
    }
    __syncthreads();
    cutIdx = (hb << 8) | s_sel;
  }
  __syncthreads();
  if (tid == 0) s_cnt = 0;
  __syncthreads();
  for (int e = tid; e < NPIX; e += 1024) {
    unsigned u = __float_as_uint(base[e]);
    bool sel = (u > V) || (u == V && (unsigned)e <= cutIdx);
    if (sel) {
      unsigned p = atomicAdd(&s_cnt, 1u);
      if (p < KPTS)
        keys[p] = ((unsigned long long)u << 16) |
                  (unsigned long long)(0xFFFFu - (unsigned)e);
    }
  }
  __syncthreads();
  // bitonic sort descending (key = value bits : inverted index)
  for (unsigned k = 2; k <= KPTS; k <<= 1) {
    for (unsigned j = k >> 1; j > 0; j >>= 1) {
      __syncthreads();
      for (unsigned t = (unsigned)tid; t < KPTS; t += 1024) {
        unsigned ixj = t ^ j;
        if (ixj > t) {
          unsigned long long a = keys[t], bb = keys[ixj];
          bool descSeg = ((t & k) == 0);
          bool sw = descSeg ? (a < bb) : (a > bb);
          if (sw) { keys[t] = bb; keys[ixj] = a; }
        }
      }
    }
  }
  __syncthreads();
  if (tid < KPTS) {
    unsigned long long kk = keys[tid];
    topkIdx[slot * KPTS + tid] = (int)(0xFFFFu - (unsigned)(kk & 0xFFFFu));
  }
}

// ---------------------------------------------------------------------------
// Kernel 3: gather descriptors at keypoints, build GEMM operands.
// Full-res feats: direct gather (align_corners makes bilinear exact).
// Half-res feats: true bilinear. Block = 192 threads (one per channel).
// img==0 -> A0 = d*w0, A1 = d*w1, N1 = d/(1e-6+||d||)
// img==1 -> D2 = d,    N2 = d/(1e-6+||d||)
// Also writes probs and landmark outputs.
// ---------------------------------------------------------------------------
__global__ __launch_bounds__(CDIM) void gather_kernel(
    const float* __restrict__ heat1, const float* __restrict__ heat2,
    const float* __restrict__ f1a, const float* __restrict__ f1b,
    const float* __restrict__ f2a, const float* __restrict__ f2b,
    const float* __restrict__ fcw, const int* __restrict__ topkIdx,
    float* __restrict__ A0, float* __restrict__ A1, float* __restrict__ N1,
    float* __restrict__ D2, float* __restrict__ N2,
    float* __restrict__ s1n, float* __restrict__ s2n,
    float* __restrict__ out) {
  int k = blockIdx.x, b = blockIdx.y, img = blockIdx.z;
  int c = threadIdx.x;
  int slot = img * 2 + b;
  int idx = topkIdx[slot * KPTS + k];
  int row = idx >> 8, col = idx & 255;
  const float* fa = img ? f2a : f1a;
  const float* fb = img ? f2b : f1b;
  float d;
  if (c < C1DIM) {
    d = fa[(((size_t)b * C1DIM + c) * HH + row) * WW + col];
  } else {
    int cc = c - C1DIM;
    float ixf = col * (127.0f / 255.0f);
    float iyf = row * (127.0f / 255.0f);
    float x0f = floorf(ixf), y0f = floorf(iyf);
    float wx = ixf - x0f, wy = iyf - y0f;
    int x0 = (int)x0f, y0 = (int)y0f;
    int x1 = (x0 + 1 < W2) ? x0 + 1 : W2 - 1;
    int y1 = (y0 + 1 < H2) ? y0 + 1 : H2 - 1;
    const float* p = fb + ((size_t)b * C2DIM + cc) * (H2 * W2);
    float v00 = p[y0 * W2 + x0], v01 = p[y0 * W2 + x1];
    float v10 = p[y1 * W2 + x0], v11 = p[y1 * W2 + x1];
    d = v00 * (1.0f - wx) * (1.0f - wy) + v01 * wx * (1.0f - wy) +
        v10 * (1.0f - wx) * wy + v11 * wx * wy;
  }
  __shared__ float sm[CDIM];
  __shared__ float s_inv, s_sq;
  sm[c] = d * d;
  __syncthreads();
  if (c < 96) sm[c] += sm[c + 96]; __syncthreads();
  if (c < 48) sm[c] += sm[c + 48]; __syncthreads();
  if (c < 24) sm[c] += sm[c + 24]; __syncthreads();
  if (c < 12) sm[c] += sm[c + 12]; __syncthreads();
  if (c < 6)  sm[c] += sm[c + 6];  __syncthreads();
  if (c < 3)  sm[c] += sm[c + 3];  __syncthreads();
  if (c == 0) {
    float sq = sm[0] + sm[1] + sm[2];
    s_sq = sq;
    s_inv = 1.0f / (1e-6f + sqrtf(sq));
  }
  __syncthreads();
  float inv = s_inv;
  size_t o = ((size_t)b * KPTS + k) * CDIM + c;
  if (img == 0) {
    A0[o] = d * fcw[c];
    A1[o] = d * fcw[CDIM + c];
    N1[o] = d * inv;
  } else {
    D2[o] = d;
    N2[o] = d * inv;
  }
  if (c == 0) {
    float sn = s_sq * inv * inv;   // ||n||^2 (with epsilon)
    (img == 0 ? s1n : s2n)[b * KPTS + k] = sn;
    const float* heat = img ? heat2 : heat1;
    float hv = 1.0f / (1.0f + __expf(-heat[(size_t)b * NPIX + idx]));
    size_t pBase = img ? PROBS2_OFF : PROBS1_OFF;
    size_t lBase = img ? LM2_OFF : LM1_OFF;
    out[pBase + (size_t)b * KPTS + k] = hv;
    float gx = col * (2.0f / 255.0f) - 1.0f;
    float gy = row * (2.0f / 255.0f) - 1.0f;
    out[lBase + ((size_t)b * KPTS + k) * 2 + 0] = gx;
    out[lBase + ((size_t)b * KPTS + k) * 2 + 1] = gy;
  }
}

// ---------------------------------------------------------------------------
// Kernel 4: fused triple fp32 WMMA GEMM (M=N=512, K=192) + epilogue.
//   score0 = A0 @ D2^T + b0 ; score1 = A1 @ D2^T + b1
//   norm   = sqrt(max(0, s1[i] + s2[j] - 2 * (N1 @ N2^T)))
// One wave per 16x16 tile, V_WMMA_F32_16X16X4_F32, 48 k-steps.
// A frag: lane(0..15)->row i0+lane, K={k0,k0+1}; lane(16..31)->K={k0+2,k0+3}.
// B frag (4x16 = D2^T tile): identical addressing on the [j,c] matrix.
// ---------------------------------------------------------------------------
__global__ __launch_bounds__(256) void match_gemm_kernel(
    const float* __restrict__ A0, const float* __restrict__ A1,
    const float* __restrict__ N1, const float* __restrict__ D2,
    const float* __restrict__ N2, const float* __restrict__ s1n,
    const float* __restrict__ s2n, const float* __restrict__ fcb,
    float* __restrict__ out) {
  int b = blockIdx.y;
  int wave = threadIdx.x >> 5;       // wave32
  int lane = threadIdx.x & 31;
  int tile = blockIdx.x * 8 + wave;  // 0..1023
  int ti = tile >> 5, tj = tile & 31;
  int i0 = ti << 4, j0 = tj << 4;

  size_t matBase = (size_t)b * KPTS * CDIM;
  const float* a0p = A0 + matBase;
  const float* a1p = A1 + matBase;
  const float* n1p = N1 + matBase;
  const float* d2p = D2 + matBase;
  const float* n2p = N2 + matBase;

  int rsel = lane & 15;
  int ksel = (lane >> 4) << 1;
  size_t offA = (size_t)(i0 + rsel) * CDIM + ksel;
  size_t offB = (size_t)(j0 + rsel) * CDIM + ksel;

  v8f acc0 = {0.f, 0.f, 0.f, 0.f, 0.f, 0.f, 0.f, 0.f};
  v8f acc1 = {0.f, 0.f, 0.f, 0.f, 0.f, 0.f, 0.f, 0.f};
  v8f accd = {0.f, 0.f, 0.f, 0.f, 0.f, 0.f, 0.f, 0.f};

  for (int k0 = 0; k0 < CDIM; k0 += 4) {
    v2f fa0 = *(const v2f*)(a0p + offA + k0);
    v2f fa1 = *(const v2f*)(a1p + offA + k0);
    v2f fn1 = *(const v2f*)(n1p + offA + k0);
    v2f fd2 = *(const v2f*)(d2p + offB + k0);
    v2f fn2 = *(const v2f*)(n2p + offB + k0);
    acc0 = __builtin_amdgcn_wmma_f32_16x16x4_f32(false, fa0, false, fd2,
                                                 (short)0, acc0, false, false);
    acc1 = __builtin_amdgcn_wmma_f32_16x16x4_f32(false, fa1, false, fd2,
                                                 (short)0, acc1, false, false);
    accd = __builtin_amdgcn_wmma_f32_16x16x4_f32(false, fn1, false, fn2,
                                                 (short)0, accd, false, false);
  }

  float b0 = fcb[0], b1 = fcb[1];
  int M0 = (lane >> 4) << 3;        // C/D layout: lanes>=16 hold M+8
  int N = lane & 15;
  int gj = j0 + N;
  float s2 = s2n[b * KPTS + gj];
  float* scoreOut = out + SCORE_OFF;
  float* normOut = out + NORM_OFF;
#pragma unroll
  for (int r = 0; r < 8; ++r) {
    int gi = i0 + M0 + r;
    size_t e = ((size_t)b * KPTS + gi) * KPTS + gj;
    scoreOut[e * 2 + 0] = acc0[r] + b0;
    scoreOut[e * 2 + 1] = acc1[r] + b1;
    float s1 = s1n[b * KPTS + gi];
    normOut[e] = sqrtf(fmaxf(s1 + s2 - 2.0f * accd[r], 0.0f));
  }
}

// ---------------------------------------------------------------------------
extern "C" void kernel_launch(void* const* d_in, const int* in_sizes, int n_in,
                              void* d_out, int out_size, void* d_ws,
                              size_t ws_size, hipStream_t stream) {
  (void)in_sizes; (void)n_in; (void)out_size; (void)ws_size;
  const float* heat1 = (const float*)d_in[0];
  const float* heat2 = (const float*)d_in[1];
  const float* f1a = (const float*)d_in[2];
  const float* f1b = (const float*)d_in[3];
  const float* f2a = (const float*)d_in[4];
  const float* f2b = (const float*)d_in[5];
  const float* fcw = (const float*)d_in[6];
  const float* fcb = (const float*)d_in[7];
  float* out = (float*)d_out;

  float* ws = (float*)d_ws;
  size_t off = 0;
  float* nms = ws + off;          off += (size_t)NSLOTS * NPIX;     // 256K floats
  int* topkIdx = (int*)(ws + off); off += (size_t)NSLOTS * KPTS;    // 2K ints
  float* A0 = ws + off;           off += (size_t)2 * KPTS * CDIM;
  float* A1 = ws + off;           off += (size_t)2 * KPTS * CDIM;
  float* N1 = ws + off;           off += (size_t)2 * KPTS * CDIM;
  float* D2 = ws + off;           off += (size_t)2 * KPTS * CDIM;
  float* N2 = ws + off;           off += (size_t)2 * KPTS * CDIM;
  float* s1n = ws + off;          off += (size_t)2 * KPTS;
  float* s2n = ws + off;          off += (size_t)2 * KPTS;

  nms_kernel<<<dim3(1024, NSLOTS), 64, 0, stream>>>(heat1, heat2, nms);
  topk_kernel<<<NSLOTS, 1024, 0, stream>>>(nms, topkIdx);
  gather_kernel<<<dim3(KPTS, 2, 2), CDIM, 0, stream>>>(
      heat1, heat2, f1a, f1b, f2a, f2b, fcw, topkIdx, A0, A1, N1, D2, N2, s1n,
      s2n, out);
  match_gemm_kernel<<<dim3(128, 2), 256, 0, stream>>>(A0, A1, N1, D2, N2, s1n,
                                                      s2n, fcb, out);
}